// GDINOMultiScaleDeformableAttention_4209067950216
// MI455X (gfx1250) — compile-verified
//
#include <hip/hip_runtime.h>
#include <hip/hip_bf16.h>

typedef __attribute__((ext_vector_type(2))) float v2f;
typedef __attribute__((ext_vector_type(8))) float v8f;

#define C_DIM  256
#define NHD    8      // heads
#define NLV    4      // levels
#define NPT    4      // points
#define HD     32     // head dim
#define BSZ    2      // batch
#define NV_TOT 13821  // sum of H*W over levels
#define NQ     NV_TOT
#define MROWS  (NQ * BSZ)   // 27642

// -------------------------------------------------------------------------
// Generic fp32 WMMA GEMM: C[M,N] = A[M,K] @ B[K,N] + bias[N]
// A row-major (lda=K), B row-major (ldb=N).
// mode 0: write C row-major (ldc=N)
// mode 1: scatter for value projection: row m -> (v=m/2, b=m&1),
//         col n -> (h=n/32, d=n&31), dest[((b*NHD+h)*NV_TOT+v)*HD+d]
// One wave computes a 16x64 tile (4 WMMA accumulators share the A fragment).
// 8 waves / block. K stepped by 4 with V_WMMA_F32_16X16X4_F32.
// -------------------------------------------------------------------------
__global__ __launch_bounds__(256)
void gemm16_kernel(const float* __restrict__ A,
                   const float* __restrict__ B,
                   const float* __restrict__ bias,
                   float* __restrict__ Cout,
                   int M, int N, int K, int mode)
{
    const int lane = threadIdx.x & 31;
    const int wave = threadIdx.x >> 5;
    const int colTiles = N >> 6;                 // 64-wide wave tiles
    const int rowTiles = (M + 15) >> 4;
    const int gw = blockIdx.x * 8 + wave;        // global wave id (wave-uniform)
    if (gw >= rowTiles * colTiles) return;       // whole-wave exit: EXEC stays full
    const int rowTile = gw / colTiles;
    const int colTile = gw % colTiles;
    const int rowBase = rowTile << 4;
    const int colBase = colTile << 6;

    const int half = lane >> 4;                  // 0: lanes 0-15, 1: lanes 16-31
    const int mrow = lane & 15;                  // A row within tile
    const int ncol = lane & 15;                  // B/C col within 16-tile

    int ar = rowBase + mrow;
    if (ar >= M) ar = M - 1;                     // clamp load row; store is guarded
    const float* Arow = A + (size_t)ar * K;

    v8f acc0 = {}, acc1 = {}, acc2 = {}, acc3 = {};

#pragma unroll 4
    for (int k = 0; k < K; k += 4) {
        const int k2 = k + 2 * half;             // this half-wave's K pair
        // A 16x4 fragment: VGPR0 = A[m][k2], VGPR1 = A[m][k2+1]
        const float2 av = *(const float2*)(Arow + k2);
        v2f a; a.x = av.x; a.y = av.y;

        // B 4x16 fragments: VGPR0 = B[k2][n], VGPR1 = B[k2+1][n]
        const float* B0 = B + (size_t)k2 * N + colBase + ncol;
        const float* B1 = B0 + N;
        v2f b0; b0.x = B0[0];  b0.y = B1[0];
        v2f b1; b1.x = B0[16]; b1.y = B1[16];
        v2f b2; b2.x = B0[32]; b2.y = B1[32];
        v2f b3; b3.x = B0[48]; b3.y = B1[48];

        acc0 = __builtin_amdgcn_wmma_f32_16x16x4_f32(false, a, false, b0, (short)0, acc0, false, false);
        acc1 = __builtin_amdgcn_wmma_f32_16x16x4_f32(false, a, false, b1, (short)0, acc1, false, false);
        acc2 = __builtin_amdgcn_wmma_f32_16x16x4_f32(false, a, false, b2, (short)0, acc2, false, false);
        acc3 = __builtin_amdgcn_wmma_f32_16x16x4_f32(false, a, false, b3, (short)0, acc3, false, false);
    }

    // Epilogue. C layout: VGPR r holds M = rowBase + r + 8*half, N = colBase + t*16 + ncol
    v8f accs[4] = {acc0, acc1, acc2, acc3};
#pragma unroll
    for (int t = 0; t < 4; ++t) {
        const int n  = colBase + t * 16 + ncol;
        const float bv = bias[n];
#pragma unroll
        for (int rr = 0; rr < 8; ++rr) {
            const int m = rowBase + rr + 8 * half;
            if (m >= M) continue;
            const float val = accs[t][rr] + bv;
            if (mode == 0) {
                Cout[(size_t)m * N + n] = val;
            } else {
                const int v = m >> 1, b = m & 1;      // value row = v*BSZ + b
                const int h = n >> 5, d = n & 31;
                Cout[(((size_t)b * NHD + h) * NV_TOT + v) * HD + d] = val;
            }
        }
    }
}

// -------------------------------------------------------------------------
// Deformable sampling: one block per (q,b) row r=q*BSZ+b, one wave per head.
// Lane = channel d in [0,32). Softmax weights are lane-uniform scalars.
// Matches torch grid_sample bilinear, align_corners=False, zero padding.
// -------------------------------------------------------------------------
__global__ __launch_bounds__(256)
void msda_sample_kernel(const float* __restrict__ offbuf,   // [MROWS][256]
                        const float* __restrict__ attnbuf,  // [MROWS][128]
                        const float* __restrict__ val_t,    // [BSZ][NHD][NV_TOT][HD]
                        const float* __restrict__ rp,       // [BSZ][NQ][NLV][2]
                        const int*   __restrict__ shapes,   // [NLV][2] (H,W)
                        const int*   __restrict__ lsi,      // [NLV]
                        float* __restrict__ mid)            // [MROWS][256]
{
    const int r    = blockIdx.x;          // r = q*BSZ + b
    const int h    = threadIdx.x >> 5;
    const int lane = threadIdx.x & 31;
    const int b    = r & 1;
    const int q    = r >> 1;

    // softmax over 16 logits for this head (redundant per lane; trivial)
    const float* al = attnbuf + (size_t)r * (NHD * NLV * NPT) + h * (NLV * NPT);
    float w[16];
    float mx = -3.0e38f;
#pragma unroll
    for (int i = 0; i < 16; ++i) { w[i] = al[i]; mx = fmaxf(mx, w[i]); }
    float s = 0.f;
#pragma unroll
    for (int i = 0; i < 16; ++i) { w[i] = __expf(w[i] - mx); s += w[i]; }
    const float inv = 1.0f / s;

    float acc = 0.f;
    const float* offr = offbuf + (size_t)r * C_DIM + h * (NLV * NPT * 2);

    for (int l = 0; l < NLV; ++l) {
        const int Hl = shapes[2 * l], Wl = shapes[2 * l + 1];
        const int start = lsi[l];
        const float fW = (float)Wl, fH = (float)Hl;
        const float rx = rp[(((size_t)b * NQ + q) * NLV + l) * 2 + 0];
        const float ry = rp[(((size_t)b * NQ + q) * NLV + l) * 2 + 1];
        const float* vh = val_t + (((size_t)b * NHD + h) * NV_TOT + start) * HD;

#pragma unroll
        for (int p = 0; p < NPT; ++p) {
            const float ox = offr[(l * NPT + p) * 2 + 0];
            const float oy = offr[(l * NPT + p) * 2 + 1];
            // loc = ref + off/[W,H];  x = loc_x*W - 0.5
            const float x = (rx + ox / fW) * fW - 0.5f;
            const float y = (ry + oy / fH) * fH - 0.5f;
            const float x0f = floorf(x), y0f = floorf(y);
            const float dx = x - x0f,  dy = y - y0f;
            const int x0 = (int)x0f,   y0 = (int)y0f;
            const float aw = w[l * NPT + p] * inv;

            const int   xs[2] = { x0, x0 + 1 };
            const int   ys[2] = { y0, y0 + 1 };
            const float wx[2] = { 1.0f - dx, dx };
            const float wy[2] = { 1.0f - dy, dy };
#pragma unroll
            for (int cy = 0; cy < 2; ++cy) {
#pragma unroll
                for (int cx = 0; cx < 2; ++cx) {
                    const int xi = xs[cx], yi = ys[cy];
                    const bool valid = (xi >= 0) & (xi < Wl) & (yi >= 0) & (yi < Hl);
                    const int xc = xi < 0 ? 0 : (xi >= Wl ? Wl - 1 : xi);
                    const int yc = yi < 0 ? 0 : (yi >= Hl ? Hl - 1 : yi);
                    const float v = vh[((size_t)yc * Wl + xc) * HD + lane];
                    acc += valid ? v * (wx[cx] * wy[cy] * aw) : 0.0f;
                }
            }
        }
    }
    mid[(size_t)r * C_DIM + h * HD + lane] = acc;
}

// -------------------------------------------------------------------------
extern "C" void kernel_launch(void* const* d_in, const int* in_sizes, int n_in,
                              void* d_out, int out_size, void* d_ws, size_t ws_size,
                              hipStream_t stream)
{
    const float* query  = (const float*)d_in[0];   // (nq, bs, C)
    const float* value  = (const float*)d_in[1];   // (NV, bs, C)
    const float* rp     = (const float*)d_in[2];   // (bs, nq, NL, 2)
    const int*   shapes = (const int*)d_in[3];     // (NL, 2)
    const int*   lsi    = (const int*)d_in[4];     // (NL,)
    const float* W_off  = (const float*)d_in[5];
    const float* b_off  = (const float*)d_in[6];
    const float* W_attn = (const float*)d_in[7];
    const float* b_attn = (const float*)d_in[8];
    const float* W_val  = (const float*)d_in[9];
    const float* b_val  = (const float*)d_in[10];
    const float* W_out  = (const float*)d_in[11];
    const float* b_out  = (const float*)d_in[12];
    float* out = (float*)d_out;

    // workspace carve-up (floats)
    float* ws      = (float*)d_ws;
    float* val_t   = ws;                                             // BSZ*NHD*NV_TOT*HD
    float* offbuf  = val_t  + (size_t)BSZ * NHD * NV_TOT * HD;       // MROWS*256
    float* attnbuf = offbuf + (size_t)MROWS * C_DIM;                 // MROWS*128
    float* mid     = attnbuf + (size_t)MROWS * (NHD * NLV * NPT);    // MROWS*256

    const int rowTiles = (MROWS + 15) / 16;   // 1728
    auto blocksFor = [&](int N) {
        int tiles = rowTiles * (N / 64);
        return (tiles + 7) / 8;
    };

    dim3 blk(256);
    // 1) value projection -> val_t (scattered to (b,h,v,d))
    gemm16_kernel<<<blocksFor(256), blk, 0, stream>>>(value, W_val, b_val, val_t,
                                                      MROWS, 256, 256, 1);
    // 2) sampling offsets
    gemm16_kernel<<<blocksFor(256), blk, 0, stream>>>(query, W_off, b_off, offbuf,
                                                      MROWS, 256, 256, 0);
    // 3) attention logits
    gemm16_kernel<<<blocksFor(128), blk, 0, stream>>>(query, W_attn, b_attn, attnbuf,
                                                      MROWS, 128, 256, 0);
    // 4) softmax + bilinear deformable sampling
    msda_sample_kernel<<<MROWS, blk, 0, stream>>>(offbuf, attnbuf, val_t, rp,
                                                  shapes, lsi, mid);
    // 5) output projection straight into d_out (row r = q*bs + b)
    gemm16_kernel<<<blocksFor(256), blk, 0, stream>>>(mid, W_out, b_out, out,
                                                      MROWS, 256, 256, 0);
}